// BiFPTreeLSTM_35940286333323
// MI455X (gfx1250) — compile-verified
//
#include <hip/hip_runtime.h>
#include <cstddef>

// ---------------- problem constants ----------------
#define NN      8192
#define KD      512      // IN == M == 512
#define MD      512
#define FIVE_M  2560
#define MAXL    64       // depth bound; random recursive tree depth ~ e*ln(N) ~ 25

typedef float v2f __attribute__((ext_vector_type(2)));
typedef float v8f __attribute__((ext_vector_type(8)));

__device__ __forceinline__ float sigm(float x) { return 1.0f / (1.0f + expf(-x)); }

// ---------------- zero init (ws is poisoned by harness) ----------------
__global__ void zero_init(float* __restrict__ acc3, size_t n_acc3,
                          float* __restrict__ c_sent, float* __restrict__ h_sent,
                          float* __restrict__ px_sent) {
    size_t idx = (size_t)blockIdx.x * blockDim.x + threadIdx.x;
    size_t stride = (size_t)gridDim.x * blockDim.x;
    for (size_t i = idx; i < n_acc3; i += stride) acc3[i] = 0.0f;
    if (idx < MD)     { c_sent[idx] = 0.0f; h_sent[idx] = 0.0f; }
    if (idx < FIVE_M) { px_sent[idx] = 0.0f; }
}

// ---------------- level assignment (sequential, O(N), trivial) ----------------
__global__ void levels_kernel(const long long* __restrict__ parent,
                              int* __restrict__ depth, int* __restrict__ order,
                              int* __restrict__ lvl_off) {
    if (blockIdx.x != 0 || threadIdx.x != 0) return;
    int cnt[MAXL];
    for (int d = 0; d < MAXL; ++d) cnt[d] = 0;
    for (int i = 0; i < NN; ++i) {
        int d;
        if (i == 0) d = 0;
        else {
            long long p = parent[i];          // parent[i] < i for i > 0
            d = depth[p] + 1;
            if (d >= MAXL) d = MAXL - 1;
        }
        depth[i] = d;
        cnt[d]++;
    }
    int acc = 0;
    for (int d = 0; d < MAXL; ++d) { lvl_off[d] = acc; acc += cnt[d]; }
    lvl_off[MAXL] = acc;
    int cur[MAXL];
    for (int d = 0; d < MAXL; ++d) cur[d] = lvl_off[d];
    for (int i = 0; i < NN; ++i) order[cur[depth[i]]++] = i;
}

// ---------------- WMMA f32 GEMM: C[N x 2560] = X[N x 512] * W[2560 x 512]^T + b --
// Block: 256 thr = 8 waves; block tile 16 rows x 256 cols; wave tile 16x32
// (two 16x16 accumulators sharing one A fragment -> 2 WMMA per A ds_load).
// Uses V_WMMA_F32_16X16X4_F32 (f32 A/B, matches reference precision).
__global__ __launch_bounds__(256) void wmma_gemm_xt(const float* __restrict__ X,
                                                    const float* __restrict__ W,
                                                    const float* __restrict__ bias,
                                                    float* __restrict__ C) {
    __shared__ float shX[16][KD + 4];
    const int tid  = threadIdx.x;
    const int wave = tid >> 5;
    const int lane = tid & 31;
    const int row0 = blockIdx.y * 16;
    const int col0 = blockIdx.x * 256 + wave * 32;

    // stage 16x512 X tile into LDS (float4 loads)
    for (int v = tid; v < 16 * (KD / 4); v += 256) {
        const int r  = v / (KD / 4);
        const int c4 = v % (KD / 4);
        const float4 x4 = ((const float4*)(X + (size_t)(row0 + r) * KD))[c4];
        shX[r][c4 * 4 + 0] = x4.x; shX[r][c4 * 4 + 1] = x4.y;
        shX[r][c4 * 4 + 2] = x4.z; shX[r][c4 * 4 + 3] = x4.w;
    }
    __syncthreads();

    // A 16x4 frag: lanes 0-15 hold K={k0,k0+1}, lanes 16-31 hold K={k0+2,k0+3}
    const int m     = lane & 15;
    const int khalf = (lane >> 4) * 2;
    const int n0    = col0 + (lane & 15);
    const float* wrow0 = W + (size_t)n0 * KD;          // B[k][n] = W[n][k]
    const float* wrow1 = wrow0 + (size_t)16 * KD;      // +16 cols = +32KB (imm offset)

    v8f acc0 = {}, acc1 = {};
#pragma unroll 4
    for (int k0 = 0; k0 < KD; k0 += 4) {
        v2f a, b0, b1;
        a[0]  = shX[m][k0 + khalf];
        a[1]  = shX[m][k0 + khalf + 1];
        b0[0] = wrow0[k0 + khalf];
        b0[1] = wrow0[k0 + khalf + 1];
        b1[0] = wrow1[k0 + khalf];
        b1[1] = wrow1[k0 + khalf + 1];
        acc0 = __builtin_amdgcn_wmma_f32_16x16x4_f32(false, a, false, b0,
                                                     (short)0, acc0, false, false);
        acc1 = __builtin_amdgcn_wmma_f32_16x16x4_f32(false, a, false, b1,
                                                     (short)0, acc1, false, false);
    }
    const float bn0 = bias[n0];
    const float bn1 = bias[n0 + 16];
    const int rbase = (lane < 16) ? 0 : 8;   // D layout: VGPR r -> M=r / M=r+8
    float* crow = C + (size_t)(row0 + rbase) * FIVE_M + n0;  // const stride folds to imm
#pragma unroll
    for (int r = 0; r < 8; ++r) {
        crow[(size_t)r * FIVE_M]      = acc0[r] + bn0;
        crow[(size_t)r * FIVE_M + 16] = acc1[r] + bn1;
    }
}

// ---------------- child-sum level kernel (bottom-up) ----------------
// One node per workgroup iteration; nodes within a level are independent.
__global__ __launch_bounds__(256) void childsum_level(
        int lvl, const int* __restrict__ lvl_off, const int* __restrict__ order,
        const long long* __restrict__ parent, const float* __restrict__ px,
        const float* __restrict__ Wio, const float* __restrict__ bio,
        const float* __restrict__ Wfz, const float* __restrict__ bfz,
        const float* __restrict__ Wum, const float* __restrict__ bum,
        float* __restrict__ acc_h, float* __restrict__ acc_fc,
        float* __restrict__ acc_zc, float* __restrict__ out0) {
    __shared__ float sh_h[MD], sh_z[MD], sh_c[MD], sh_hn[MD];
    const int base  = lvl_off[lvl];
    const int count = lvl_off[lvl + 1] - base;
    const int tid   = threadIdx.x;

    for (int j = blockIdx.x; j < count; j += gridDim.x) {
        const int i = order[base + j];
        for (int mm = tid; mm < MD; mm += 256) {
            sh_h[mm] = acc_h [(size_t)i * MD + mm];
            sh_z[mm] = acc_zc[(size_t)i * MD + mm];
        }
        __syncthreads();
        const float* pxi = px + (size_t)i * FIVE_M;   // chunks: i,f,o,z,u
#pragma unroll
        for (int mm = 0; mm < 2; ++mm) {
            const int  mo  = tid + mm * 256;
            const float4* wi4 = (const float4*)(Wio + (size_t)mo * MD);
            const float4* wo4 = (const float4*)(Wio + (size_t)(MD + mo) * MD);
            const float4* wu4 = (const float4*)(Wum + (size_t)mo * MD);
            float di = 0.f, dz = 0.f, du = 0.f;
            for (int k4 = 0; k4 < MD / 4; ++k4) {
                const float4 a = wi4[k4], b = wo4[k4], u4 = wu4[k4];
                const int k = k4 * 4;
                di += a.x * sh_h[k] + a.y * sh_h[k+1] + a.z * sh_h[k+2] + a.w * sh_h[k+3];
                dz += b.x * sh_h[k] + b.y * sh_h[k+1] + b.z * sh_h[k+2] + b.w * sh_h[k+3];
                du += u4.x * sh_z[k] + u4.y * sh_z[k+1] + u4.z * sh_z[k+2] + u4.w * sh_z[k+3];
            }
            const float ig = sigm(pxi[mo]            + di + bio[mo]);
            const float og = sigm(pxi[2 * MD + mo]   + dz + bio[MD + mo]);
            const float u  = tanhf(pxi[4 * MD + mo]  + du + bum[mo]);
            const float c  = ig * u + acc_fc[(size_t)i * MD + mo];
            sh_c[mo]  = c;
            sh_hn[mo] = og * tanhf(c);
        }
        __syncthreads();
        const long long p = parent[i];                // root -> sentinel NN (zero row)
        const float* pxp  = px + (size_t)p * FIVE_M;
#pragma unroll
        for (int mm = 0; mm < 2; ++mm) {
            const int  mo  = tid + mm * 256;
            const float4* wf4 = (const float4*)(Wfz + (size_t)mo * MD);
            const float4* wz4 = (const float4*)(Wfz + (size_t)(MD + mo) * MD);
            float df = 0.f, dzz = 0.f;
            for (int k4 = 0; k4 < MD / 4; ++k4) {
                const float4 a = wf4[k4], b = wz4[k4];
                const int k = k4 * 4;
                df  += a.x * sh_hn[k] + a.y * sh_hn[k+1] + a.z * sh_hn[k+2] + a.w * sh_hn[k+3];
                dzz += b.x * sh_hn[k] + b.y * sh_hn[k+1] + b.z * sh_hn[k+2] + b.w * sh_hn[k+3];
            }
            const float f = sigm(pxp[MD + mo]     + df  + bfz[mo]);
            const float z = sigm(pxp[3 * MD + mo] + dzz + bfz[MD + mo]);
            const float c = sh_c[mo], h = sh_hn[mo];
            atomicAdd(&acc_h [(size_t)p * MD + mo], h);
            atomicAdd(&acc_fc[(size_t)p * MD + mo], f * c);
            atomicAdd(&acc_zc[(size_t)p * MD + mo], z * tanhf(c));
            if (i == 0) out0[mo] = h;                 // root hidden -> out[0:M]
        }
        __syncthreads();
    }
}

// ---------------- chain level kernel (top-down) ----------------
__global__ __launch_bounds__(256) void chain_level(
        int lvl, const int* __restrict__ lvl_off, const int* __restrict__ order,
        const long long* __restrict__ parent, const float* __restrict__ qx,
        const float* __restrict__ Wh, const float* __restrict__ bh,
        const float* __restrict__ Wum, const float* __restrict__ bum,
        float* __restrict__ c_all, float* __restrict__ h_all) {
    __shared__ float sh_ph[MD], sh_pc[MD], sh_t[MD];
    const int base  = lvl_off[lvl];
    const int count = lvl_off[lvl + 1] - base;
    const int tid   = threadIdx.x;

    for (int j = blockIdx.x; j < count; j += gridDim.x) {
        const int i       = order[base + j];
        const long long p = parent[i];                // node 0 -> sentinel NN (zeros)
        for (int mm = tid; mm < MD; mm += 256) {
            sh_ph[mm] = h_all[(size_t)p * MD + mm];
            sh_pc[mm] = c_all[(size_t)p * MD + mm];
        }
        __syncthreads();
        const float* qxi = qx + (size_t)i * FIVE_M;   // chunks: i,o,f,z,u
        float igv[2], ogv[2], fgv[2];
#pragma unroll
        for (int mm = 0; mm < 2; ++mm) {
            const int mo = tid + mm * 256;
            const float4* wi4 = (const float4*)(Wh + (size_t)mo * MD);
            const float4* wo4 = (const float4*)(Wh + (size_t)(MD + mo) * MD);
            const float4* wf4 = (const float4*)(Wh + (size_t)(2 * MD + mo) * MD);
            const float4* wz4 = (const float4*)(Wh + (size_t)(3 * MD + mo) * MD);
            float gi = 0.f, go = 0.f, gf = 0.f, gz = 0.f;
            for (int k4 = 0; k4 < MD / 4; ++k4) {
                const float4 a = wi4[k4], b = wo4[k4], c = wf4[k4], d = wz4[k4];
                const int k = k4 * 4;
                const float h0 = sh_ph[k], h1 = sh_ph[k+1], h2 = sh_ph[k+2], h3 = sh_ph[k+3];
                gi += a.x*h0 + a.y*h1 + a.z*h2 + a.w*h3;
                go += b.x*h0 + b.y*h1 + b.z*h2 + b.w*h3;
                gf += c.x*h0 + c.y*h1 + c.z*h2 + c.w*h3;
                gz += d.x*h0 + d.y*h1 + d.z*h2 + d.w*h3;
            }
            igv[mm] = sigm(qxi[mo]            + gi + bh[mo]);
            ogv[mm] = sigm(qxi[MD + mo]       + go + bh[MD + mo]);
            fgv[mm] = sigm(qxi[2 * MD + mo]   + gf + bh[2 * MD + mo]);
            const float zg = sigm(qxi[3 * MD + mo] + gz + bh[3 * MD + mo]);
            sh_t[mo] = zg * tanhf(sh_pc[mo]);
        }
        __syncthreads();
#pragma unroll
        for (int mm = 0; mm < 2; ++mm) {
            const int mo = tid + mm * 256;
            const float4* wu4 = (const float4*)(Wum + (size_t)mo * MD);
            float du = 0.f;
            for (int k4 = 0; k4 < MD / 4; ++k4) {
                const float4 a = wu4[k4];
                const int k = k4 * 4;
                du += a.x * sh_t[k] + a.y * sh_t[k+1] + a.z * sh_t[k+2] + a.w * sh_t[k+3];
            }
            const float u = tanhf(qxi[4 * MD + mo] + du + bum[mo]);
            const float c = igv[mm] * u + fgv[mm] * sh_pc[mo];
            const float h = ogv[mm] * tanhf(c);
            c_all[(size_t)i * MD + mo] = c;
            h_all[(size_t)i * MD + mo] = h;
        }
        __syncthreads();
    }
}

// ---------------- finalize: out[M:2M] = colmax over h_all[:N] ----------------
__global__ void finalize_max(const float* __restrict__ h_all, float* __restrict__ out) {
    const int mo = blockIdx.x * blockDim.x + threadIdx.x;
    if (mo >= MD) return;
    float mx = -INFINITY;
    for (int n = 0; n < NN; ++n) mx = fmaxf(mx, h_all[(size_t)n * MD + mo]);
    out[MD + mo] = mx;
}

// ---------------- host-side launch sequence ----------------
extern "C" void kernel_launch(void* const* d_in, const int* in_sizes, int n_in,
                              void* d_out, int out_size, void* d_ws, size_t ws_size,
                              hipStream_t stream) {
    const float*     inputs = (const float*)d_in[0];
    const long long* parent = (const long long*)d_in[1];
    const float *cs_Wx = (const float*)d_in[2],  *cs_bx  = (const float*)d_in[3];
    const float *cs_Wio= (const float*)d_in[4],  *cs_bio = (const float*)d_in[5];
    const float *cs_Wfz= (const float*)d_in[6],  *cs_bfz = (const float*)d_in[7];
    const float *cs_Wum= (const float*)d_in[8],  *cs_bum = (const float*)d_in[9];
    const float *ch_Wx = (const float*)d_in[10], *ch_bx  = (const float*)d_in[11];
    const float *ch_Wh = (const float*)d_in[12], *ch_bh  = (const float*)d_in[13];
    const float *ch_Wum= (const float*)d_in[14], *ch_bum = (const float*)d_in[15];
    float* out = (float*)d_out;

    // workspace layout (floats)
    float* f = (float*)d_ws;
    size_t o = 0;
    float* px     = f + o; o += (size_t)(NN + 1) * FIVE_M;  // sentinel row NN (zero)
    float* qx     = f + o; o += (size_t)NN * FIVE_M;
    float* acc_h  = f + o; o += (size_t)(NN + 1) * MD;      // acc_h/fc/zc contiguous
    float* acc_fc = f + o; o += (size_t)(NN + 1) * MD;
    float* acc_zc = f + o; o += (size_t)(NN + 1) * MD;
    float* c_all  = f + o; o += (size_t)(NN + 1) * MD;
    float* h_all  = f + o; o += (size_t)(NN + 1) * MD;
    int* ip      = (int*)(f + o);
    int* depth   = ip;
    int* order   = ip + NN;
    int* lvl_off = ip + 2 * NN;                             // MAXL+1 entries

    // 1) zero accumulators + sentinel rows (ws is poisoned by harness)
    zero_init<<<512, 256, 0, stream>>>(acc_h, (size_t)3 * (NN + 1) * MD,
                                       c_all + (size_t)NN * MD,
                                       h_all + (size_t)NN * MD,
                                       px + (size_t)NN * FIVE_M);
    // 2) level bucketing
    levels_kernel<<<1, 1, 0, stream>>>(parent, depth, order, lvl_off);
    // 3) input projections via f32 WMMA GEMM (block tile 16 x 256)
    dim3 ggrid(FIVE_M / 256, NN / 16);
    wmma_gemm_xt<<<ggrid, 256, 0, stream>>>(inputs, cs_Wx, cs_bx, px);
    wmma_gemm_xt<<<ggrid, 256, 0, stream>>>(inputs, ch_Wx, ch_bx, qx);
    // 4) bottom-up child-sum: deepest level first
    for (int lvl = MAXL - 1; lvl >= 0; --lvl)
        childsum_level<<<512, 256, 0, stream>>>(lvl, lvl_off, order, parent, px,
                                                cs_Wio, cs_bio, cs_Wfz, cs_bfz,
                                                cs_Wum, cs_bum,
                                                acc_h, acc_fc, acc_zc, out);
    // 5) top-down chain: level 0 upward
    for (int lvl = 0; lvl < MAXL; ++lvl)
        chain_level<<<512, 256, 0, stream>>>(lvl, lvl_off, order, parent, qx,
                                             ch_Wh, ch_bh, ch_Wum, ch_bum,
                                             c_all, h_all);
    // 6) column-max over all chain hiddens -> out[M:2M]
    finalize_max<<<2, 256, 0, stream>>>(h_all, out);
}